// FAConvGNN_86938728005836
// MI455X (gfx1250) — compile-verified
//
#include <hip/hip_runtime.h>
#include <math.h>

#define D    128
#define OUTD 128
#define EPS  0.1f

typedef __attribute__((ext_vector_type(2))) float v2f;
typedef __attribute__((ext_vector_type(8))) float v8f;

// Native CDNA5 no-return fp32 atomic add (device scope handled by L2).
// Avoids any chance of a CAS expansion for the 82M scatter-adds.
__device__ __forceinline__ void atomic_add_f32_native(float* p, float v)
{
    asm volatile("global_atomic_add_f32 %0, %1, off"
                 :
                 : "v"(p), "v"(v)
                 : "memory");
}

// ---------------------------------------------------------------------------
// Kernel 1: per-node attention logits (two GEMVs) + agg = EPS * x_0
// one wave32 per node; lane l owns columns [4l, 4l+3]
// ---------------------------------------------------------------------------
__global__ void __launch_bounds__(256)
faconv_node_prep(const float* __restrict__ x,
                 const float* __restrict__ x0,
                 const float* __restrict__ attl,
                 const float* __restrict__ attr,
                 float* __restrict__ alpha_l,
                 float* __restrict__ alpha_r,
                 float* __restrict__ agg,
                 int N)
{
    int wave = (int)((blockIdx.x * blockDim.x + threadIdx.x) >> 5);
    int lane = (int)(threadIdx.x & 31);
    if (wave >= N) return;

    const float4* xr  = (const float4*)(x  + (size_t)wave * D);
    const float4* x0r = (const float4*)(x0 + (size_t)wave * D);

    float4 xv = xr[lane];
    float4 al = ((const float4*)attl)[lane];
    float4 ar = ((const float4*)attr)[lane];

    float pl = xv.x * al.x + xv.y * al.y + xv.z * al.z + xv.w * al.w;
    float pr = xv.x * ar.x + xv.y * ar.y + xv.z * ar.z + xv.w * ar.w;

    #pragma unroll
    for (int off = 16; off > 0; off >>= 1) {
        pl += __shfl_xor(pl, off, 32);
        pr += __shfl_xor(pr, off, 32);
    }
    if (lane == 0) {
        alpha_l[wave] = pl;
        alpha_r[wave] = pr;
    }

    float4 v0 = x0r[lane];
    float4 a;
    a.x = EPS * v0.x; a.y = EPS * v0.y; a.z = EPS * v0.z; a.w = EPS * v0.w;
    ((float4*)(agg + (size_t)wave * D))[lane] = a;
}

// ---------------------------------------------------------------------------
// Kernel 2: per-edge gather + scale + scatter-add (native fp32 atomics)
// one wave32 per edge; lane l owns columns [4l, 4l+3]
// ---------------------------------------------------------------------------
__global__ void __launch_bounds__(256)
faconv_edge(const float* __restrict__ x,
            const int*   __restrict__ src,
            const int*   __restrict__ dst,
            const float* __restrict__ ew,
            const float* __restrict__ alpha_l,
            const float* __restrict__ alpha_r,
            float* __restrict__ agg,
            int E)
{
    int wave = (int)((blockIdx.x * blockDim.x + threadIdx.x) >> 5);
    int lane = (int)(threadIdx.x & 31);
    if (wave >= E) return;

    int s = src[wave];
    int d = dst[wave];

    const float* xs = x + (size_t)s * D;
    __builtin_prefetch(xs + lane * 4, 0, 0);   // global_prefetch_b8

    float coeff = tanhf(alpha_l[s] + alpha_r[d]) * ew[wave];

    float4 xv = ((const float4*)xs)[lane];
    float* ag = agg + (size_t)d * D + lane * 4;

    atomic_add_f32_native(ag + 0, coeff * xv.x);
    atomic_add_f32_native(ag + 1, coeff * xv.y);
    atomic_add_f32_native(ag + 2, coeff * xv.z);
    atomic_add_f32_native(ag + 3, coeff * xv.w);
}

// ---------------------------------------------------------------------------
// Kernel 3: one-block repack of W into WMMA-B-fragment-friendly layout:
//   Wp[pair][o] = (W[o][2*pair], W[o][2*pair+1])   as float2
// so a lane's B fragment (b.x = B[kk][n], b.y = B[kk+1][n]) is ONE b64 load.
// ---------------------------------------------------------------------------
__global__ void __launch_bounds__(256)
faconv_wt(const float* __restrict__ W, float* __restrict__ Wp)
{
    for (int idx = (int)threadIdx.x; idx < OUTD * D; idx += (int)blockDim.x) {
        int o = idx >> 7;     // row of W   (output channel)
        int k = idx & 127;    // col of W   (input channel)
        // pair = k>>1, slot = k&1
        Wp[(size_t)(k >> 1) * (OUTD * 2) + o * 2 + (k & 1)] = W[idx];
    }
}

// ---------------------------------------------------------------------------
// Kernel 4: out = relu(agg) @ W^T + b   via V_WMMA_F32_16X16X4_F32
// block = 256 threads = 8 waves; block owns a 16-row strip of agg (in LDS),
// wave w computes output columns [16w, 16w+15].
//
// A fragment (16x4 f32, per ISA layout): lanes 0-15 -> M=lane, v0=K0,v1=K1;
//                                        lanes 16-31 -> M=lane-16, v0=K2,v1=K3.
// B fragment (4x16 f32, mirrored):       v0 = B[K0|K2][n], v1 = B[K1|K3][n], n=lane&15.
// C/D (16x16 f32): VGPR v -> row v (lanes 0-15) / row v+8 (lanes 16-31), col = lane&15.
// ---------------------------------------------------------------------------
__global__ void __launch_bounds__(256)
faconv_gemm(const float* __restrict__ agg,
            const float* __restrict__ Wp,     // paired layout, see faconv_wt
            const float* __restrict__ bias,
            float* __restrict__ out,
            int N)
{
    __shared__ float lA[16][D + 4];

    int r0 = (int)blockIdx.x * 16;
    if (r0 >= N) return;

    // cooperative load of the 16xD strip with relu applied
    for (int idx = (int)threadIdx.x; idx < 16 * (D / 4); idx += (int)blockDim.x) {
        int m  = idx >> 5;        // 0..15   (32 float4 per row)
        int c4 = idx & 31;
        int r  = r0 + m;
        float4 v = ((const float4*)(agg + (size_t)(r < N ? r : N - 1) * D))[c4];
        lA[m][c4 * 4 + 0] = fmaxf(v.x, 0.0f);
        lA[m][c4 * 4 + 1] = fmaxf(v.y, 0.0f);
        lA[m][c4 * 4 + 2] = fmaxf(v.z, 0.0f);
        lA[m][c4 * 4 + 3] = fmaxf(v.w, 0.0f);
    }
    __syncthreads();

    int wave  = (int)(threadIdx.x >> 5);
    int lane  = (int)(threadIdx.x & 31);
    int c0    = wave * 16;            // first output column of this wave
    int n     = lane & 15;            // M for A-frag, N for B/C/D-frags
    int khalf = (lane >> 4) * 2;      // 0 for lanes 0-15, 2 for lanes 16-31

    const float2* Wp2 = (const float2*)Wp;   // Wp2[pair * OUTD + o] = (B[2p][o], B[2p+1][o])

    v8f c = {};
    #pragma unroll
    for (int k = 0; k < D; k += 4) {
        int kk = k + khalf;                       // even: k or k+2
        v2f a, b;
        a.x = lA[n][kk];
        a.y = lA[n][kk + 1];
        float2 bv = Wp2[(size_t)(kk >> 1) * OUTD + c0 + n];   // one b64 load
        b.x = bv.x;
        b.y = bv.y;
        // (neg_a, A, neg_b, B, c_mod, C, reuse_a, reuse_b)
        c = __builtin_amdgcn_wmma_f32_16x16x4_f32(false, a, false, b,
                                                  (short)0, c, false, false);
    }

    float bv    = bias[c0 + n];
    int   mbase = (lane >> 4) * 8;
    #pragma unroll
    for (int v = 0; v < 8; ++v) {
        int r = r0 + mbase + v;
        if (r < N)
            out[(size_t)r * OUTD + c0 + n] = c[v] + bv;
    }
}

// ---------------------------------------------------------------------------
// Launch
// inputs: 0:x 1:x_0 2:edge_weight 3:att_l 4:att_r 5:W 6:b 7:edge_index
// ---------------------------------------------------------------------------
extern "C" void kernel_launch(void* const* d_in, const int* in_sizes, int n_in,
                              void* d_out, int out_size, void* d_ws, size_t ws_size,
                              hipStream_t stream)
{
    const float* x    = (const float*)d_in[0];
    const float* x0   = (const float*)d_in[1];
    const float* ew   = (const float*)d_in[2];
    const float* attl = (const float*)d_in[3];
    const float* attr = (const float*)d_in[4];
    const float* W    = (const float*)d_in[5];
    const float* b    = (const float*)d_in[6];
    const int*   ei   = (const int*)d_in[7];

    int N = in_sizes[0] / D;    // 100000
    int E = in_sizes[2];        // 640000
    const int* src = ei;
    const int* dst = ei + E;

    float* wsf     = (float*)d_ws;
    float* alpha_l = wsf;
    float* alpha_r = wsf + (size_t)N;
    float* Wp      = wsf + (size_t)2 * N;
    float* agg     = wsf + (size_t)2 * N + (size_t)D * OUTD;

    float* out = (float*)d_out;

    // 1) node logits + agg init (8 waves per block -> 8 nodes per block)
    {
        int blocks = (N + 7) / 8;
        faconv_node_prep<<<blocks, 256, 0, stream>>>(x, x0, attl, attr,
                                                     alpha_l, alpha_r, agg, N);
    }
    // 2) W repack (tiny, single block)
    faconv_wt<<<1, 256, 0, stream>>>(W, Wp);

    // 3) edge scatter (8 edges per block)
    {
        int blocks = (E + 7) / 8;
        faconv_edge<<<blocks, 256, 0, stream>>>(x, src, dst, ew,
                                                alpha_l, alpha_r, agg, E);
    }
    // 4) fused relu + GEMM + bias (16 rows per block)
    {
        int blocks = (N + 15) / 16;
        faconv_gemm<<<blocks, 256, 0, stream>>>(agg, Wp, b, out, N);
    }
}